// LightSE_49709951484604
// MI455X (gfx1250) — compile-verified
//
#include <hip/hip_runtime.h>
#include <hip/hip_bf16.h>

// SE block: pool -> (256->32 relu) -> (32->256 sigmoid) -> rescale
// B=32, CH=256, MID=32, H=W=64 (plane = 4096 floats = 16 KiB)

typedef __attribute__((ext_vector_type(2))) float v2f;
typedef __attribute__((ext_vector_type(4))) float v4f;
typedef __attribute__((ext_vector_type(8))) float v8f;

#define CH   256
#define MID  32
#define BATCH 32
#define PLANE 4096   // 64*64

// ---------------------------------------------------------------------------
// Kernel 1: global average pool. One block per (b,c) plane.
// ---------------------------------------------------------------------------
__global__ __launch_bounds__(256) void se_pool(const float* __restrict__ x,
                                               float* __restrict__ s) {
    const int plane = blockIdx.x;                       // b*CH + c
    const v4f* xp = (const v4f*)(x + (size_t)plane * PLANE);
    const int tid = threadIdx.x;

    float sum = 0.f;
#pragma unroll
    for (int j = 0; j < 4; ++j) {
        v4f v = xp[tid + j * 256];
        sum += v.x + v.y + v.z + v.w;
    }
    // wave32 reduction
#pragma unroll
    for (int off = 16; off > 0; off >>= 1)
        sum += __shfl_down(sum, off, 32);

    __shared__ float partial[8];
    const int wave = tid >> 5;
    if ((tid & 31) == 0) partial[wave] = sum;
    __syncthreads();
    if (tid == 0) {
        float t = 0.f;
#pragma unroll
        for (int w = 0; w < 8; ++w) t += partial[w];
        s[plane] = t * (1.0f / (float)PLANE);
    }
}

// ---------------------------------------------------------------------------
// Kernel 2: gate = sigmoid(relu(s @ w1^T + b1) @ w2^T + b2), via WMMA f32.
// Single block, 256 threads = 8 waves.
//
// WMMA fragment layout (CDNA5 ISA 7.12.2, 32-bit operands, wave32):
//   A 16x4 : lanes 0-15  hold row M=lane,  v0=K0, v1=K1
//            lanes 16-31 hold row M=lane-16, v0=K2, v1=K3
//   B 4x16 : lanes 0-15  hold col N=lane,  v0=K0, v1=K1
//            lanes 16-31 hold col N=lane-16, v0=K2, v1=K3
//   C/D 16x16 : lanes 0-15 vgpr j = (M=j, N=lane); lanes 16-31 = (M=j+8, N=lane-16)
// ---------------------------------------------------------------------------
__global__ __launch_bounds__(256) void se_gate(const float* __restrict__ s,   // [32,256]
                                               const float* __restrict__ w1,  // [32,256]
                                               const float* __restrict__ b1,  // [32]
                                               const float* __restrict__ w2,  // [256,32]
                                               const float* __restrict__ b2,  // [256]
                                               float* __restrict__ g) {       // [32,256]
    __shared__ float hbuf[BATCH * MID];   // h matrix, 32x32

    const int tid  = threadIdx.x;
    const int wave = tid >> 5;
    const int lane = tid & 31;
    const int half = lane >> 4;    // 0: lanes 0-15, 1: lanes 16-31
    const int l16  = lane & 15;

    // ---- GEMM1: h[32x32] = relu(s[32x256] @ w1^T + b1), K=256 ----
    if (wave < 4) {
        const int tm = wave & 1;       // M tile (batch rows)
        const int tn = wave >> 1;      // N tile (mid channels)
        v8f acc = {};
        const float* arow = s  + (size_t)(tm * 16 + l16) * CH + 2 * half;
        const float* brow = w1 + (size_t)(tn * 16 + l16) * CH + 2 * half;
        for (int k0 = 0; k0 < CH; k0 += 4) {
            v2f a = *(const v2f*)(arow + k0);
            v2f b = *(const v2f*)(brow + k0);
            acc = __builtin_amdgcn_wmma_f32_16x16x4_f32(
                false, a, false, b, (short)0, acc, false, false);
        }
        const float bias = b1[tn * 16 + l16];
#pragma unroll
        for (int j = 0; j < 8; ++j) {
            float v = acc[j] + bias;
            v = v > 0.f ? v : 0.f;                      // relu
            const int M = tm * 16 + j + 8 * half;       // batch row
            const int N = tn * 16 + l16;                // mid channel
            hbuf[M * MID + N] = v;
        }
    }
    __syncthreads();

    // ---- GEMM2: g = sigmoid(h[32x32] @ w2^T + b2), K=32 ----
    for (int t = wave; t < 32; t += 8) {
        const int tm = t & 1;          // M tile (batch rows)
        const int tn = t >> 1;         // N tile (output channels)
        v8f acc = {};
        const float* arow = hbuf + (tm * 16 + l16) * MID + 2 * half;
        const float* brow = w2   + (size_t)(tn * 16 + l16) * MID + 2 * half;
#pragma unroll
        for (int k0 = 0; k0 < MID; k0 += 4) {
            v2f a = *(const v2f*)(arow + k0);
            v2f b = *(const v2f*)(brow + k0);
            acc = __builtin_amdgcn_wmma_f32_16x16x4_f32(
                false, a, false, b, (short)0, acc, false, false);
        }
        const float bias = b2[tn * 16 + l16];
#pragma unroll
        for (int j = 0; j < 8; ++j) {
            const float v  = acc[j] + bias;
            const int  bb  = tm * 16 + j + 8 * half;    // batch row
            const int  cc  = tn * 16 + l16;             // out channel
            g[bb * CH + cc] = 1.0f / (1.0f + __expf(-v));
        }
    }
}

// ---------------------------------------------------------------------------
// Kernel 3: out = x * g (broadcast over plane). x should hit in the 192 MB L2
// (resident from se_pool's read); out uses non-temporal stores so the output
// stream does not evict x before it is consumed.
// ---------------------------------------------------------------------------
__global__ __launch_bounds__(256) void se_scale(const float* __restrict__ x,
                                                const float* __restrict__ g,
                                                float* __restrict__ out) {
    const int plane = blockIdx.x;
    const float gv = g[plane];                          // uniform -> scalar load
    const v4f* xp = (const v4f*)(x + (size_t)plane * PLANE);
    v4f* op = (v4f*)(out + (size_t)plane * PLANE);
    const int tid = threadIdx.x;
#pragma unroll
    for (int j = 0; j < 4; ++j) {
        v4f v = xp[tid + j * 256];
        v *= gv;
        __builtin_nontemporal_store(v, &op[tid + j * 256]);
    }
}

// ---------------------------------------------------------------------------
extern "C" void kernel_launch(void* const* d_in, const int* in_sizes, int n_in,
                              void* d_out, int out_size, void* d_ws, size_t ws_size,
                              hipStream_t stream) {
    const float* x  = (const float*)d_in[0];   // [32,256,64,64]
    const float* w1 = (const float*)d_in[1];   // [32,256]
    const float* b1 = (const float*)d_in[2];   // [32]
    const float* w2 = (const float*)d_in[3];   // [256,32]
    const float* b2 = (const float*)d_in[4];   // [256]
    float* out = (float*)d_out;

    float* ws_s = (float*)d_ws;                // pooled means [32*256]
    float* ws_g = ws_s + BATCH * CH;           // gates        [32*256]

    const int nplanes = BATCH * CH;            // 8192

    se_pool <<<nplanes, 256, 0, stream>>>(x, ws_s);
    se_gate <<<1,       256, 0, stream>>>(ws_s, w1, b1, w2, b2, ws_g);
    se_scale<<<nplanes, 256, 0, stream>>>(x, ws_g, out);
}